// ConvR_37306085933469
// MI455X (gfx1250) — compile-verified
//
#include <hip/hip_runtime.h>
#include <hip/hip_bf16.h>
#include <math.h>

typedef __attribute__((ext_vector_type(2))) float v2f;
typedef __attribute__((ext_vector_type(4))) float v4f;
typedef __attribute__((ext_vector_type(8))) float v8f;
typedef __attribute__((ext_vector_type(4))) unsigned int v4u;
typedef __attribute__((ext_vector_type(4))) int v4i;
typedef __attribute__((ext_vector_type(8))) int v8i;

#define B_      512
#define NUM_ENT 100000
#define NUM_REL 1000
#define EDIM    200
#define CH      32
#define OH      8
#define OW      18
#define OHW     (OH*OW)          // 144
#define FEAT    (CH*OHW)         // 4608
#define EPS_    1e-5f

// workspace layout (float offsets)
#define WS_SCALE0   0
#define WS_SHIFT0   1
#define WS_SUM1     8      // 32 floats
#define WS_SQ1      40     // 32 floats
#define WS_SCALE1   72     // 32 floats
#define WS_SHIFT1   104    // 32 floats
#define WS_Y        256                      // B*FEAT = 2359296
#define WS_ACT      (WS_Y + B_*FEAT)         // B*FEAT
#define WS_Z        (WS_ACT + B_*FEAT)       // B*EDIM
#define WS_H        (WS_Z + B_*EDIM)         // B*EDIM

// ---------------------------------------------------------------- zero accums
__global__ void k_zero(float* ws) {
    int t = threadIdx.x;
    if (t < 64) ws[WS_SUM1 + t] = 0.0f;   // covers sum1[32] + sq1[32]
}

// ---------------------------------------------------------------- bn0 stats
__global__ void k_bn0(const int* e1, const float* emb_e,
                      const float* g0, const float* b0, float* ws) {
    __shared__ float ssum[256], ssq[256];
    int t = threadIdx.x;
    float s = 0.f, q = 0.f;
    for (int i = t; i < B_ * EDIM; i += 256) {
        int b = i / EDIM, k = i - b * EDIM;
        float x = emb_e[e1[b] * EDIM + k];
        s += x; q += x * x;
    }
    ssum[t] = s; ssq[t] = q;
    __syncthreads();
    for (int o = 128; o > 0; o >>= 1) {
        if (t < o) { ssum[t] += ssum[t + o]; ssq[t] += ssq[t + o]; }
        __syncthreads();
    }
    if (t == 0) {
        float inv = 1.0f / (float)(B_ * EDIM);
        float mean = ssum[0] * inv;
        float var  = ssq[0] * inv - mean * mean;
        float rstd = rsqrtf(var + EPS_);
        float sc = g0[0] * rstd;
        ws[WS_SCALE0] = sc;
        ws[WS_SHIFT0] = b0[0] - mean * sc;
    }
}

// ---------------------------------------------------------------- conv per sample
__global__ void k_conv(const int* e1, const int* rel, const float* emb_e,
                       const float* conv_w, const float* conv_b, float* ws) {
    __shared__ float xs[EDIM];    // normalized 10x20 image
    __shared__ float wm[CH * 9];  // gathered 32x3x3 weights
    int b = blockIdx.x;
    int t = threadIdx.x;          // 128 threads
    float sc = ws[WS_SCALE0], sh = ws[WS_SHIFT0];
    int r = rel[b];
    for (int i = t; i < EDIM; i += 128)
        xs[i] = emb_e[e1[b] * EDIM + i] * sc + sh;
    for (int i = t; i < CH * 9; i += 128)
        wm[i] = conv_w[r * (CH * 9) + i];
    __syncthreads();

    int c = t & 31;
    float bias = conv_b[r * CH + c];
    float lsum = 0.f, lsq = 0.f;
    float* yout = ws + WS_Y + (size_t)b * FEAT + c * OHW;
    for (int p = (t >> 5); p < OHW; p += 4) {
        int oh = p / OW, ow = p - oh * OW;
        float acc = bias;
        #pragma unroll
        for (int kh = 0; kh < 3; ++kh)
            #pragma unroll
            for (int kw = 0; kw < 3; ++kw)
                acc += wm[c * 9 + kh * 3 + kw] * xs[(oh + kh) * 20 + (ow + kw)];
        yout[p] = acc;
        lsum += acc; lsq += acc * acc;
    }
    atomicAdd(&ws[WS_SUM1 + c], lsum);
    atomicAdd(&ws[WS_SQ1  + c], lsq);
}

// ---------------------------------------------------------------- bn1 finalize
__global__ void k_bn1fin(const float* g1, const float* b1, float* ws) {
    int c = threadIdx.x;
    if (c < CH) {
        float inv = 1.0f / (float)(B_ * OHW);
        float mean = ws[WS_SUM1 + c] * inv;
        float var  = ws[WS_SQ1 + c] * inv - mean * mean;
        float rstd = rsqrtf(var + EPS_);
        float sc = g1[c] * rstd;
        ws[WS_SCALE1 + c] = sc;
        ws[WS_SHIFT1 + c] = b1[c] - mean * sc;
    }
}

// ---------------------------------------------------------------- bn1 apply + relu
__global__ void k_act(float* ws) {
    int i = blockIdx.x * 256 + threadIdx.x;
    if (i < B_ * FEAT) {
        int c = (i % FEAT) / OHW;
        float v = ws[WS_Y + i] * ws[WS_SCALE1 + c] + ws[WS_SHIFT1 + c];
        ws[WS_ACT + i] = v > 0.f ? v : 0.f;
    }
}

// ---------------------------------------------------------------- fc GEMM (WMMA f32)
// z[512 x 200] = act[512 x 4608] @ fc_w^T + fc_b
// No divergent loads: OOB columns are clamped (their results are never stored).
__global__ void k_fc(const float* fc_w, const float* fc_b, float* ws) {
    const float* act = ws + WS_ACT;
    float* z = ws + WS_Z;
    int lane = threadIdx.x & 31;
    int ln = lane & 15;
    int hi = (lane >= 16) ? 1 : 0;
    int m0 = blockIdx.y * 16;
    int n0 = blockIdx.x * 16;
    int f  = n0 + ln;
    int fc_ = (f < EDIM) ? f : (EDIM - 1);   // clamp; junk columns never stored
    int row = m0 + ln;
    int s0 = hi ? 2 : 0;

    v8f acc = {};
    const float* arow = act + (size_t)row * FEAT + s0;
    const float* brow = fc_w + (size_t)fc_ * FEAT + s0;
    #pragma unroll 4
    for (int kk = 0; kk < FEAT; kk += 4) {
        v2f a  = *(const v2f*)(arow + kk);
        v2f bb = *(const v2f*)(brow + kk);
        acc = __builtin_amdgcn_wmma_f32_16x16x4_f32(
            false, a, false, bb, (short)0, acc, false, false);
    }
    if (f < EDIM) {
        float fb = fc_b[f];
        #pragma unroll
        for (int j = 0; j < 8; ++j) {
            int mr = m0 + j + (hi ? 8 : 0);
            z[mr * EDIM + f] = acc[j] + fb;
        }
    }
}

// ---------------------------------------------------------------- bn2 + relu -> h
__global__ void k_bn2(const float* g2, const float* b2, float* ws) {
    int f = threadIdx.x;
    if (f >= EDIM) return;
    const float* z = ws + WS_Z;
    float* h = ws + WS_H;
    float s = 0.f, q = 0.f;
    for (int b = 0; b < B_; ++b) {
        float v = z[b * EDIM + f];
        s += v; q += v * v;
    }
    float inv = 1.0f / (float)B_;
    float mean = s * inv;
    float var  = q * inv - mean * mean;
    float rstd = rsqrtf(var + EPS_);
    float sc = g2[f] * rstd;
    float sh = b2[f] - mean * sc;
    for (int b = 0; b < B_; ++b) {
        float v = z[b * EDIM + f] * sc + sh;
        h[b * EDIM + f] = v > 0.f ? v : 0.f;
    }
}

// ---------------------------------------------------------------- logits GEMM (WMMA f32) + sigmoid
// out[512 x 100000] = sigmoid(h[512 x 200] @ emb_e^T + bias_e)
// block: 128 thr (4 waves). grid: (6250 N-tiles, 8 M-blocks of 64 rows)
// B tile (16 entities x 200 k, row-major) staged via Tensor Data Mover.
// A tile (64 rows of h) staged via b128 copies, stride 204 (16B-aligned, conflict-free).
#define A_STRIDE 204
__global__ void k_logits(const float* emb_e, const float* bias_e,
                         const float* ws, float* out) {
    __shared__ float ldsB[16 * EDIM];        // [n][k] row-major, 12.8 KB
    __shared__ float ldsA[64 * A_STRIDE];    // 64 rows of h, padded
    const float* h = ws + WS_H;
    int tid = threadIdx.x;
    int n0 = blockIdx.x * 16;
    int mblk = blockIdx.y * 64;

    // --- TDM: DMA 16 rows x 200 f32 of emb_e into ldsB (wave 0 issues) ---
    if (tid < 1) {
        unsigned int lds_off = (unsigned int)(size_t)(&ldsB[0]);
        unsigned long long ga =
            (unsigned long long)(size_t)(emb_e + (size_t)n0 * EDIM);
        v4u g0;
        g0[0] = 1u;                                   // count=1
        g0[1] = lds_off;                              // lds_addr
        g0[2] = (unsigned int)ga;                     // global_addr[31:0]
        g0[3] = (unsigned int)((ga >> 32) & 0x1FFFFFFu) | (2u << 30); // +type=2
        v8i g1;
        g1[0] = 2 << 16;          // data_size = 2 (4 bytes)
        g1[1] = EDIM << 16;       // tensor_dim0 = 200 (low 16)
        g1[2] = 16 << 16;         // tensor_dim1 = 16 (low 16)
        g1[3] = EDIM << 16;       // tile_dim0 = 200
        g1[4] = 16;               // tile_dim1 = 16
        g1[5] = EDIM;             // tensor_dim0_stride = 200
        g1[6] = 0;
        g1[7] = 0;
        v4i gz4 = {};
        v8i gz8 = {};
        __builtin_amdgcn_tensor_load_to_lds(g0, g1, gz4, gz4, gz8, 0);
    }
    // --- stage 64 rows of h with b128 copies ---
    for (int i = tid; i < 64 * (EDIM / 4); i += 128) {
        int m = i / (EDIM / 4);
        int k4 = (i - m * (EDIM / 4)) * 4;
        v4f v = *(const v4f*)(h + (size_t)(mblk + m) * EDIM + k4);
        *(v4f*)(&ldsA[m * A_STRIDE + k4]) = v;
    }
    __builtin_amdgcn_s_wait_tensorcnt(0);
    __syncthreads();

    int wave = tid >> 5;
    int lane = tid & 31;
    int ln = lane & 15;
    int hi = (lane >= 16) ? 1 : 0;
    int s0 = hi ? 2 : 0;
    const float* arow = ldsA + (wave * 16 + ln) * A_STRIDE + s0;
    const float* brow = ldsB + ln * EDIM + s0;

    v8f acc = {};
    #pragma unroll 10
    for (int kk = 0; kk < EDIM; kk += 4) {
        v2f a  = *(const v2f*)(arow + kk);
        v2f bb = *(const v2f*)(brow + kk);
        acc = __builtin_amdgcn_wmma_f32_16x16x4_f32(
            false, a, false, bb, (short)0, acc, false, false);
    }

    int col = n0 + ln;
    float be = bias_e[col];
    int mbase = mblk + wave * 16 + (hi ? 8 : 0);
    #pragma unroll
    for (int j = 0; j < 8; ++j) {
        float v = acc[j] + be;
        out[(size_t)(mbase + j) * NUM_ENT + col] = 1.0f / (1.0f + __expf(-v));
    }
}

// ----------------------------------------------------------------
extern "C" void kernel_launch(void* const* d_in, const int* in_sizes, int n_in,
                              void* d_out, int out_size, void* d_ws, size_t ws_size,
                              hipStream_t stream) {
    const int*   e1     = (const int*)d_in[0];
    const int*   rel    = (const int*)d_in[1];
    const float* emb_e  = (const float*)d_in[2];
    const float* conv_w = (const float*)d_in[3];
    const float* conv_b = (const float*)d_in[4];
    const float* g0     = (const float*)d_in[5];
    const float* b0     = (const float*)d_in[6];
    const float* g1     = (const float*)d_in[7];
    const float* b1     = (const float*)d_in[8];
    const float* g2     = (const float*)d_in[9];
    const float* b2     = (const float*)d_in[10];
    const float* fc_w   = (const float*)d_in[11];
    const float* fc_b   = (const float*)d_in[12];
    const float* bias_e = (const float*)d_in[13];
    float* out = (float*)d_out;
    float* ws  = (float*)d_ws;

    k_zero<<<1, 64, 0, stream>>>(ws);
    k_bn0<<<1, 256, 0, stream>>>(e1, emb_e, g0, b0, ws);
    k_conv<<<B_, 128, 0, stream>>>(e1, rel, emb_e, conv_w, conv_b, ws);
    k_bn1fin<<<1, 32, 0, stream>>>(g1, b1, ws);
    k_act<<<(B_ * FEAT + 255) / 256, 256, 0, stream>>>(ws);
    k_fc<<<dim3((EDIM + 15) / 16, B_ / 16), 32, 0, stream>>>(fc_w, fc_b, ws);
    k_bn2<<<1, 256, 0, stream>>>(g2, b2, ws);
    k_logits<<<dim3(NUM_ENT / 16, B_ / 64), 128, 0, stream>>>(emb_e, bias_e, ws, out);
}